// adj_embedding_592705487496
// MI455X (gfx1250) — compile-verified
//
#include <hip/hip_runtime.h>

#define NN    10000
#define TOPK  32            // K_NEIGHS // 2
#define TILE  16
#define TROWS 625           // 10000 / 16 exactly
#define WPB   5             // waves per block in kernel 2 (625 = 5 * 125)

typedef float v2f __attribute__((ext_vector_type(2)));
typedef float v8f __attribute__((ext_vector_type(8)));

// ---------------------------------------------------------------------------
// Kernel 1: top-32 of emb_t (value desc, ties -> lower index, matching
// jax.lax.top_k). Produces masked vector m: m[j] = emb_t[j] if selected else 0.
// Single workgroup; all data lives in LDS (40KB << 320KB/WGP).
// ---------------------------------------------------------------------------
__global__ __launch_bounds__(1024) void topk_mask_kernel(
    const float* __restrict__ emb_t, float* __restrict__ m) {
  __shared__ float vals[NN];
  __shared__ unsigned long long red[1024];
  const int tid = threadIdx.x;

  for (int j = tid; j < NN; j += 1024) vals[j] = emb_t[j];
  __syncthreads();

  for (int it = 0; it < TOPK; ++it) {
    // Packed key: float bits in high word (valid ordering since vals >= 0),
    // ~index in low word so equal values prefer the smaller index.
    unsigned long long best = 0ull;
    for (int j = tid; j < NN; j += 1024) {
      float v = vals[j];
      if (v >= 0.0f) {
        unsigned long long key =
            ((unsigned long long)__float_as_uint(v) << 32) |
            (unsigned int)(~j);
        if (key > best) best = key;
      }
    }
    red[tid] = best;
    __syncthreads();
    for (int s = 512; s > 0; s >>= 1) {
      if (tid < s && red[tid + s] > red[tid]) red[tid] = red[tid + s];
      __syncthreads();
    }
    if (tid == 0) {
      unsigned int winIdx = ~(unsigned int)(red[0] & 0xFFFFFFFFull);
      vals[winIdx] = -1.0f;   // mark selected; real values are all >= 0
    }
    __syncthreads();
  }

  for (int j = tid; j < NN; j += 1024)
    m[j] = (vals[j] < 0.0f) ? emb_t[j] : 0.0f;
}

// ---------------------------------------------------------------------------
// Kernel 2: out[i,j] = emb_s[i] * m[j] via V_WMMA_F32_16X16X4_F32.
// One 16x16 tile per wave. A: 16x4 with only K=0 column = emb_s chunk
// (VGPR0 lanes 0-15 = K=0, lanes 16-31 = K=2 -> zero; VGPR1 = K=1/K=3 -> zero).
// B: 4x16 with only K=0 row = m chunk (same striping transposed).
// D accumulates exactly one product per element -> bit-exact f32.
// Store: VGPR r of lane L is element (row = ti*16 + r + 8*(L>=16),
// col = tj*16 + L%16) -> 8 global b32 stores, 64B row segments; neighbor
// waves (tj+1) complete the 128B lines in L2 before eviction.
// ---------------------------------------------------------------------------
__global__ __launch_bounds__(WPB * 32) void outer_topk_wmma_kernel(
    const float* __restrict__ emb_s, const float* __restrict__ m,
    float* __restrict__ out) {
  const int wave = threadIdx.x >> 5;
  const int lane = threadIdx.x & 31;
  const int half = lane >> 4;   // 0: lanes 0-15, 1: lanes 16-31
  const int l16  = lane & 15;

  const int ti = blockIdx.y;                 // tile row   [0,625)
  const int tj = blockIdx.x * WPB + wave;    // tile col   [0,625)

  v2f a, b;
  a.x = (half == 0) ? emb_s[ti * TILE + l16] : 0.0f;  // A[:,K=0]; K=2 half zero
  a.y = 0.0f;                                         // K=1 / K=3
  b.x = (half == 0) ? m[tj * TILE + l16] : 0.0f;      // B[K=0,:]; K=2 half zero
  b.y = 0.0f;                                         // K=1 / K=3

  v8f c = {};
  v8f d = __builtin_amdgcn_wmma_f32_16x16x4_f32(
      /*neg_a=*/false, a, /*neg_b=*/false, b,
      /*c_mod=*/(short)0, c, /*reuse_a=*/false, /*reuse_b=*/false);

  const int row0 = ti * TILE + half * 8;
  const int col  = tj * TILE + l16;
  float* p = out + (size_t)row0 * NN + col;
#pragma unroll
  for (int r = 0; r < 8; ++r) p[(size_t)r * NN] = d[r];
}

// ---------------------------------------------------------------------------
extern "C" void kernel_launch(void* const* d_in, const int* in_sizes, int n_in,
                              void* d_out, int out_size, void* d_ws,
                              size_t ws_size, hipStream_t stream) {
  const float* emb_s = (const float*)d_in[0];  // [10000,1]
  const float* emb_t = (const float*)d_in[1];  // [1,10000]
  float* out = (float*)d_out;                  // [10000,10000]
  float* m   = (float*)d_ws;                   // 40KB scratch: masked emb_t

  topk_mask_kernel<<<1, 1024, 0, stream>>>(emb_t, m);

  dim3 grid(TROWS / WPB, TROWS);               // 125 x 625 blocks, 5 waves each
  outer_topk_wmma_kernel<<<grid, WPB * 32, 0, stream>>>(emb_s, m, out);
}